// AdditiveAttnLayer_4947802325396
// MI455X (gfx1250) — compile-verified
//
#include <hip/hip_runtime.h>

#define N_  50000
#define NE_ 800000

typedef __attribute__((ext_vector_type(2))) float v2f;
typedef __attribute__((ext_vector_type(8))) float v8f;

#define DEV __device__ __forceinline__

// ---------------- WMMA helpers (CDNA5 f32 16x16x4) ----------------
DEV v8f wmma4(v2f a, v2f b, v8f c) {
  // 8 args: (neg_a, A, neg_b, B, c_mod, C, reuse_a, reuse_b)
  return __builtin_amdgcn_wmma_f32_16x16x4_f32(false, a, false, b, (short)0, c,
                                               false, false);
}

DEV v8f vzero() { v8f z; for (int i = 0; i < 8; ++i) z[i] = 0.f; return z; }

// A-fragment (16x4 f32): lane%16 = row, lane/16 selects k pair {k0+2h, k0+2h+1}
DEV v2f afrag_g(const float* base, int row, int ld, int k0, int lane) {
  int half = lane >> 4;
  const float* p = base + (size_t)row * ld + k0 + 2 * half;
  v2f a; a.x = p[0]; a.y = p[1]; return a;
}

// B-fragment (4x16 f32) from row-major weight W[outCol][k] (ld = K):
// B[k][n] = W[n][k] -> same lane pattern as A but row index = output column.
DEV v2f bfrag_lds(const float* w, int n0, int ld, int k0, int lane) {
  int half = lane >> 4;
  const float* p = w + (size_t)(n0 + (lane & 15)) * ld + k0 + 2 * half;
  v2f b; b.x = p[0]; b.y = p[1]; return b;
}

// ---------------- float <-> monotone uint key (for atomic max) ----------------
DEV unsigned f2key(float f) {
  unsigned b = __float_as_uint(f);
  return (b & 0x80000000u) ? ~b : (b | 0x80000000u);
}
DEV float key2f(unsigned k) {
  unsigned b = (k & 0x80000000u) ? (k & 0x7fffffffu) : ~k;
  return __uint_as_float(b);
}

// ---------------- workspace layout (float offsets) ----------------
static constexpr size_t OFF_Q     = 0;          // N*64
static constexpr size_t OFF_K     = 3200000;    // N*64
static constexpr size_t OFF_V     = 6400000;    // N*64
static constexpr size_t OFF_SCORE = 9600000;    // NE*64 (== oE == e_t)
static constexpr size_t OFF_AEX   = 60800000;   // NE*8  (a, then ex in-place)
static constexpr size_t OFF_AMAX  = 67200000;   // N*8 (uint keys)
static constexpr size_t OFF_DENOM = 67600000;   // N*8
static constexpr size_t OFF_AGGV  = 68000000;   // N*64
static constexpr size_t OFF_AGGE  = 71200000;   // N*64
static constexpr size_t OFF_NHRES = 74400000;   // N*64
static constexpr size_t OFF_H1    = 77600000;   // N*128
static constexpr size_t OFF_NH3   = 84000000;   // N*64
static constexpr size_t OFF_STATS = 87200000;   // 1024
// stats sub-layout: [0]=bn1n sum,[64]=bn1n sq,[128]=bn1e sum,[192]=bn1e sq,
// [256]=bn2 sum,[320]=bn2 sq,[384]=scaleN,[448]=shiftN,[512]=scaleE,
// [576]=shiftE,[640]=scale2,[704]=shift2

// ---------------- K0: init ----------------
__global__ void k_init(float* ws) {
  size_t i = (size_t)blockIdx.x * 256 + threadIdx.x;
  if (i < 7200000) {          // AMAX + DENOM + AGGV + AGGE (contiguous)
    ws[OFF_AMAX + i] = 0.f;   // key 0 < any real value's key
  } else if (i < 7200000 + 1024) {
    ws[OFF_STATS + (i - 7200000)] = 0.f;
  }
}

// ---------------- K1: Q,K,V projections ----------------
__global__ void k_qkv(const float* __restrict__ x,
                      const float* Wq, const float* bq,
                      const float* Wk, const float* bk,
                      const float* Wv, const float* bv,
                      float* Qb, float* Kb, float* Vb) {
  __shared__ float w[3 * 4096];
  __shared__ float bias[3 * 64];
  for (int i = threadIdx.x; i < 4096; i += 256) {
    w[i] = Wq[i]; w[4096 + i] = Wk[i]; w[8192 + i] = Wv[i];
  }
  for (int i = threadIdx.x; i < 64; i += 256) {
    bias[i] = bq[i]; bias[64 + i] = bk[i]; bias[128 + i] = bv[i];
  }
  __syncthreads();
  int wave = threadIdx.x >> 5, lane = threadIdx.x & 31;
  int m0 = (blockIdx.x * 8 + wave) * 16;
  if (m0 >= N_) return;
  bool full = (m0 + 16 <= N_);          // wave-uniform
  int row = m0 + (lane & 15); if (row >= N_) row = N_ - 1;
  v2f af[16];
#pragma unroll
  for (int kk = 0; kk < 16; ++kk) af[kk] = afrag_g(x, row, 64, kk * 4, lane);
  float* outs[3] = {Qb, Kb, Vb};
  int half = lane >> 4, col0 = lane & 15;
  for (int m = 0; m < 3; ++m) {
    for (int n0 = 0; n0 < 64; n0 += 16) {
      v8f acc = vzero();
#pragma unroll
      for (int kk = 0; kk < 16; ++kk)
        acc = wmma4(af[kk], bfrag_lds(w + m * 4096, n0, 64, kk * 4, lane), acc);
      float b = bias[m * 64 + n0 + col0];
      int col = n0 + col0;
      if (full) {
#pragma unroll
        for (int v = 0; v < 8; ++v)
          outs[m][(size_t)(m0 + v + 8 * half) * 64 + col] = acc[v] + b;
      } else {
#pragma unroll
        for (int v = 0; v < 8; ++v) {
          int r = m0 + v + 8 * half;
          if (r < N_) outs[m][(size_t)r * 64 + col] = acc[v] + b;
        }
      }
    }
  }
}

// ---------------- K2: edge pass A (E proj + score + logits + seg-max) --------
// NE is an exact multiple of 128, so no edge tail handling is needed.
__global__ void k_edgeA(const float* __restrict__ edge_attr,
                        const int* __restrict__ src, const int* __restrict__ dst,
                        const float* We, const float* be, const float* Aw,
                        const float* __restrict__ Qb, const float* __restrict__ Kb,
                        float* scoreBuf, float* aBuf, unsigned* amaxU) {
  __shared__ float w[8192];     // We 128x64
  __shared__ float bias[128];
  __shared__ float awl[64];     // Aw[d][h]
  __shared__ float estage[8 * 256];
  for (int i = threadIdx.x; i < 8192; i += 256) w[i] = We[i];
  for (int i = threadIdx.x; i < 128; i += 256) bias[i] = be[i];
  for (int i = threadIdx.x; i < 64; i += 256) awl[i] = Aw[i];
  __syncthreads();
  int wave = threadIdx.x >> 5, lane = threadIdx.x & 31;
  int e0 = (blockIdx.x * 8 + wave) * 16;
  int half = lane >> 4, col0 = lane & 15;
  int rowe = e0 + col0;
  v2f af[16];
#pragma unroll
  for (int kk = 0; kk < 16; ++kk) af[kk] = afrag_g(edge_attr, rowe, 64, kk * 4, lane);
  int le = lane >> 1;                 // edge-in-tile for elementwise phase
  int dbase = (lane & 1) * 4;         // d = dbase..dbase+3
  int ge = e0 + le;
  int s = src[ge], dn = dst[ge];
  for (int h = 0; h < 8; ++h) {
    v8f acc = vzero();
#pragma unroll
    for (int kk = 0; kk < 16; ++kk)
      acc = wmma4(af[kk], bfrag_lds(w, h * 16, 64, kk * 4, lane), acc);
    float b = bias[h * 16 + col0];
#pragma unroll
    for (int v = 0; v < 8; ++v)
      estage[wave * 256 + (v + 8 * half) * 16 + col0] = acc[v] + b;
    __syncthreads();
    const float4 kv = *(const float4*)(Kb + (size_t)s  * 64 + h * 8 + dbase);
    const float4 qv = *(const float4*)(Qb + (size_t)dn * 64 + h * 8 + dbase);
    float kq[4] = {kv.x, kv.y, kv.z, kv.w};
    float qq[4] = {qv.x, qv.y, qv.z, qv.w};
    float sc[4];
    float asum = 0.f;
#pragma unroll
    for (int i = 0; i < 4; ++i) {
      float e1 = estage[wave * 256 + le * 16 + dbase + i];
      float e2 = estage[wave * 256 + le * 16 + 8 + dbase + i];
      float s2 = e1 * e2;
      float ss = (s2 > 0.f) ? sqrtf(s2) : ((s2 < 0.f) ? -sqrtf(-s2) : 0.f);
      float val = kq[i] + qq[i] + ss;
      sc[i] = val > 0.f ? val : 0.f;
      asum += sc[i] * awl[(dbase + i) * 8 + h];
    }
    float4 sv; sv.x = sc[0]; sv.y = sc[1]; sv.z = sc[2]; sv.w = sc[3];
    *(float4*)(scoreBuf + (size_t)ge * 64 + h * 8 + dbase) = sv;
    asum += __shfl_xor(asum, 1, 32);
    if ((lane & 1) == 0) {
      float a = fminf(fmaxf(asum, -5.f), 5.f);
      aBuf[(size_t)ge * 8 + h] = a;
      atomicMax(&amaxU[(size_t)dn * 8 + h], f2key(a));
    }
    __syncthreads();
  }
}

// ---------------- K3: exp + denom ----------------
__global__ void k_softmax1(const int* __restrict__ dst, const unsigned* __restrict__ amaxU,
                           float* aEx, float* denom) {
  size_t i = (size_t)blockIdx.x * 256 + threadIdx.x;   // grid == NE*8 exactly
  int e = (int)(i >> 3), h = (int)(i & 7);
  int dn = dst[e];
  float am = key2f(amaxU[(size_t)dn * 8 + h]);
  float ex = __expf(aEx[i] - am);
  aEx[i] = ex;
  atomicAdd(&denom[(size_t)dn * 8 + h], ex);
}

// ---------------- K4: attention scatter ----------------
__global__ void k_scatter(const int* __restrict__ src, const int* __restrict__ dst,
                          const float* __restrict__ aEx, const float* __restrict__ denom,
                          const float* __restrict__ Vb, const float* __restrict__ scoreBuf,
                          float* aggV, float* aggE) {
  size_t i = (size_t)blockIdx.x * 256 + threadIdx.x;   // grid == NE*8 exactly
  int e = (int)(i >> 3), h = (int)(i & 7);
  int dn = dst[e], sn = src[e];
  float attn = aEx[i] / (denom[(size_t)dn * 8 + h] + 1e-16f);
  const float* vp = Vb + (size_t)sn * 64 + h * 8;
  const float* sp = scoreBuf + (size_t)e * 64 + h * 8;
  float* av = aggV + (size_t)dn * 64 + h * 8;
  float* ae = aggE + (size_t)dn * 64 + h * 8;
#pragma unroll
  for (int d = 0; d < 8; ++d) {
    atomicAdd(&av[d], vp[d] * attn);
    atomicAdd(&ae[d], sp[d] * attn);
  }
}

// ---------------- K5: node mix + No GEMM + residual + BN1n partials ----------
__global__ void k_node(const float* __restrict__ x, const float* __restrict__ log_deg,
                       const float* __restrict__ aggV, const float* __restrict__ aggE,
                       const float* VeRow, const float* deg_coef,
                       const float* Now, const float* Nob,
                       float* nhRes, float* bnStats) {
  __shared__ float w[4096];
  __shared__ float bias[64];
  __shared__ float ve[512];
  __shared__ float dc[128];
  __shared__ float stage[8 * 1024];
  __shared__ float csum[64], csq[64];
  for (int i = threadIdx.x; i < 4096; i += 256) w[i] = Now[i];
  for (int i = threadIdx.x; i < 512; i += 256) ve[i] = VeRow[i];
  for (int i = threadIdx.x; i < 128; i += 256) dc[i] = deg_coef[i];
  if (threadIdx.x < 64) { bias[threadIdx.x] = Nob[threadIdx.x];
                          csum[threadIdx.x] = 0.f; csq[threadIdx.x] = 0.f; }
  __syncthreads();
  int wave = threadIdx.x >> 5, lane = threadIdx.x & 31;
  int m0 = (blockIdx.x * 8 + wave) * 16;
  bool full = (m0 + 16 <= N_);
  // elementwise: VeRow einsum + degree scaler -> stage
  for (int r = 0; r < 16; ++r) {
    int n = m0 + r; int nn = (n < N_) ? n : (N_ - 1);
    float ld = log_deg[nn];
#pragma unroll
    for (int t = 0; t < 2; ++t) {
      int j = lane + t * 32;
      int h = j >> 3, c = j & 7;
      const float* gep = aggE + (size_t)nn * 64 + h * 8;
      float rv = 0.f;
#pragma unroll
      for (int d = 0; d < 8; ++d) rv += gep[d] * ve[d * 64 + h * 8 + c];
      float tmp = aggV[(size_t)nn * 64 + j] + rv;
      stage[wave * 1024 + r * 64 + j] = tmp * dc[j * 2] + tmp * ld * dc[j * 2 + 1];
    }
  }
  __syncthreads();
  int half = lane >> 4, col0 = lane & 15;
  v2f af[16];
#pragma unroll
  for (int kk = 0; kk < 16; ++kk) {
    int k = kk * 4 + 2 * half;
    af[kk].x = stage[wave * 1024 + col0 * 64 + k];
    af[kk].y = stage[wave * 1024 + col0 * 64 + k + 1];
  }
  for (int n0 = 0; n0 < 64; n0 += 16) {
    v8f acc = vzero();
#pragma unroll
    for (int kk = 0; kk < 16; ++kk)
      acc = wmma4(af[kk], bfrag_lds(w, n0, 64, kk * 4, lane), acc);
    int col = n0 + col0;
    float b = bias[col];
    float psum = 0.f, psq = 0.f;
    if (full) {
#pragma unroll
      for (int v = 0; v < 8; ++v) {
        size_t r = (size_t)(m0 + v + 8 * half);
        float val = acc[v] + b + x[r * 64 + col];
        nhRes[r * 64 + col] = val;
        psum += val; psq += val * val;
      }
    } else {
#pragma unroll
      for (int v = 0; v < 8; ++v) {
        int r = m0 + v + 8 * half;
        if (r < N_) {
          float val = acc[v] + b + x[(size_t)r * 64 + col];
          nhRes[(size_t)r * 64 + col] = val;
          psum += val; psq += val * val;
        }
      }
    }
    atomicAdd(&csum[col], psum);
    atomicAdd(&csq[col], psq);
  }
  __syncthreads();
  if (threadIdx.x < 64) {
    atomicAdd(&bnStats[threadIdx.x], csum[threadIdx.x]);
    atomicAdd(&bnStats[64 + threadIdx.x], csq[threadIdx.x]);
  }
}

// ---------------- K6: eh = edge_attr + oE@Eo^T + b, BN1e partials -----------
__global__ void k_edgeOut(const float* __restrict__ edge_attr,
                          const float* __restrict__ scoreBuf,
                          const float* Eow, const float* Eob,
                          float* outE, float* bnStats) {
  __shared__ float w[4096];
  __shared__ float bias[64];
  __shared__ float csum[64], csq[64];
  for (int i = threadIdx.x; i < 4096; i += 256) w[i] = Eow[i];
  if (threadIdx.x < 64) { bias[threadIdx.x] = Eob[threadIdx.x];
                          csum[threadIdx.x] = 0.f; csq[threadIdx.x] = 0.f; }
  __syncthreads();
  int wave = threadIdx.x >> 5, lane = threadIdx.x & 31;
  int e0 = (blockIdx.x * 8 + wave) * 16;
  int half = lane >> 4, col0 = lane & 15;
  int rowe = e0 + col0;
  v2f af[16];
#pragma unroll
  for (int kk = 0; kk < 16; ++kk) af[kk] = afrag_g(scoreBuf, rowe, 64, kk * 4, lane);
  for (int n0 = 0; n0 < 64; n0 += 16) {
    v8f acc = vzero();
#pragma unroll
    for (int kk = 0; kk < 16; ++kk)
      acc = wmma4(af[kk], bfrag_lds(w, n0, 64, kk * 4, lane), acc);
    int col = n0 + col0;
    float b = bias[col];
    float psum = 0.f, psq = 0.f;
#pragma unroll
    for (int v = 0; v < 8; ++v) {
      size_t r = (size_t)(e0 + v + 8 * half);
      float val = acc[v] + b + edge_attr[r * 64 + col];
      outE[r * 64 + col] = val;
      psum += val; psq += val * val;
    }
    atomicAdd(&csum[col], psum);
    atomicAdd(&csq[col], psq);
  }
  __syncthreads();
  if (threadIdx.x < 64) {
    atomicAdd(&bnStats[128 + threadIdx.x], csum[threadIdx.x]);
    atomicAdd(&bnStats[192 + threadIdx.x], csq[threadIdx.x]);
  }
}

// ---------------- K7: finalize BN1 (node + edge) -> scale/shift --------------
__global__ void k_finalize1(float* stats, const float* g1n, const float* b1n,
                            const float* g1e, const float* b1e) {
  int t = threadIdx.x;
  if (t < 64) {
    float mu = stats[t] / (float)N_;
    float var = stats[64 + t] / (float)N_ - mu * mu;
    float sc = rsqrtf(var + 1e-5f) * g1n[t];
    stats[384 + t] = sc;
    stats[448 + t] = b1n[t] - mu * sc;
  } else if (t < 128) {
    int j = t - 64;
    float mu = stats[128 + j] / (float)NE_;
    float var = stats[192 + j] / (float)NE_ - mu * mu;
    float sc = rsqrtf(var + 1e-5f) * g1e[j];
    stats[512 + j] = sc;
    stats[576 + j] = b1e[j] - mu * sc;
  }
}

// ---------------- K8: apply BN1e to edge output in place ---------------------
__global__ void k_bn_edge(float* outE, const float* __restrict__ stats) {
  size_t i = (size_t)blockIdx.x * 256 + threadIdx.x;   // grid == NE*16 exactly
  size_t base = i * 4;
  int col = (int)(base & 63);
  float4 v = *(float4*)(outE + base);
  v.x = v.x * stats[512 + col] + stats[576 + col];
  v.y = v.y * stats[512 + col + 1] + stats[576 + col + 1];
  v.z = v.z * stats[512 + col + 2] + stats[576 + col + 2];
  v.w = v.w * stats[512 + col + 3] + stats[576 + col + 3];
  *(float4*)(outE + base) = v;
}

// ---------------- K9a: h1 = relu(bn1(nhRes) @ W1^T + b1f) --------------------
__global__ void k_ffn1(const float* __restrict__ nhRes, const float* __restrict__ stats,
                       const float* W1, const float* b1f, float* h1Buf) {
  __shared__ float w[8192];     // 128x64
  __shared__ float bias[128];
  __shared__ float scl[64], shl[64];
  for (int i = threadIdx.x; i < 8192; i += 256) w[i] = W1[i];
  for (int i = threadIdx.x; i < 128; i += 256) bias[i] = b1f[i];
  if (threadIdx.x < 64) { scl[threadIdx.x] = stats[384 + threadIdx.x];
                          shl[threadIdx.x] = stats[448 + threadIdx.x]; }
  __syncthreads();
  int wave = threadIdx.x >> 5, lane = threadIdx.x & 31;
  int m0 = (blockIdx.x * 8 + wave) * 16;
  if (m0 >= N_) return;
  bool full = (m0 + 16 <= N_);
  int half = lane >> 4, col0 = lane & 15;
  int row = m0 + col0; if (row >= N_) row = N_ - 1;
  v2f af[16];
#pragma unroll
  for (int kk = 0; kk < 16; ++kk) {
    int k = kk * 4 + 2 * half;
    af[kk].x = nhRes[(size_t)row * 64 + k]     * scl[k]     + shl[k];
    af[kk].y = nhRes[(size_t)row * 64 + k + 1] * scl[k + 1] + shl[k + 1];
  }
  for (int n0 = 0; n0 < 128; n0 += 16) {
    v8f acc = vzero();
#pragma unroll
    for (int kk = 0; kk < 16; ++kk)
      acc = wmma4(af[kk], bfrag_lds(w, n0, 64, kk * 4, lane), acc);
    int col = n0 + col0;
    float b = bias[col];
    if (full) {
#pragma unroll
      for (int v = 0; v < 8; ++v) {
        float val = acc[v] + b;
        h1Buf[(size_t)(m0 + v + 8 * half) * 128 + col] = val > 0.f ? val : 0.f;
      }
    } else {
#pragma unroll
      for (int v = 0; v < 8; ++v) {
        int r = m0 + v + 8 * half;
        if (r < N_) {
          float val = acc[v] + b;
          h1Buf[(size_t)r * 128 + col] = val > 0.f ? val : 0.f;
        }
      }
    }
  }
}

// ---------------- K9b: nh3 = bn1(nhRes) + h1 @ W2^T + b2f, BN2 partials ------
__global__ void k_ffn2(const float* __restrict__ h1Buf, const float* __restrict__ nhRes,
                       const float* __restrict__ stats, const float* W2, const float* b2f,
                       float* nh3, float* bnStats) {
  __shared__ float w[8192];     // 64x128
  __shared__ float bias[64];
  __shared__ float scl[64], shl[64];
  __shared__ float csum[64], csq[64];
  for (int i = threadIdx.x; i < 8192; i += 256) w[i] = W2[i];
  if (threadIdx.x < 64) {
    bias[threadIdx.x] = b2f[threadIdx.x];
    scl[threadIdx.x] = stats[384 + threadIdx.x];
    shl[threadIdx.x] = stats[448 + threadIdx.x];
    csum[threadIdx.x] = 0.f; csq[threadIdx.x] = 0.f;
  }
  __syncthreads();
  int wave = threadIdx.x >> 5, lane = threadIdx.x & 31;
  int m0 = (blockIdx.x * 8 + wave) * 16;
  bool full = (m0 + 16 <= N_);
  int half = lane >> 4, col0 = lane & 15;
  int row = m0 + col0; if (row >= N_) row = N_ - 1;
  v2f af[32];
#pragma unroll
  for (int kk = 0; kk < 32; ++kk) af[kk] = afrag_g(h1Buf, row, 128, kk * 4, lane);
  for (int n0 = 0; n0 < 64; n0 += 16) {
    v8f acc = vzero();
#pragma unroll
    for (int kk = 0; kk < 32; ++kk)
      acc = wmma4(af[kk], bfrag_lds(w, n0, 128, kk * 4, lane), acc);
    int col = n0 + col0;
    float b = bias[col];
    float psum = 0.f, psq = 0.f;
    if (full) {
#pragma unroll
      for (int v = 0; v < 8; ++v) {
        size_t r = (size_t)(m0 + v + 8 * half);
        float resid = nhRes[r * 64 + col] * scl[col] + shl[col];
        float val = acc[v] + b + resid;
        nh3[r * 64 + col] = val;
        psum += val; psq += val * val;
      }
    } else {
#pragma unroll
      for (int v = 0; v < 8; ++v) {
        int r = m0 + v + 8 * half;
        if (r < N_) {
          float resid = nhRes[(size_t)r * 64 + col] * scl[col] + shl[col];
          float val = acc[v] + b + resid;
          nh3[(size_t)r * 64 + col] = val;
          psum += val; psq += val * val;
        }
      }
    }
    atomicAdd(&csum[col], psum);
    atomicAdd(&csq[col], psq);
  }
  __syncthreads();
  if (threadIdx.x < 64) {
    atomicAdd(&bnStats[256 + threadIdx.x], csum[threadIdx.x]);
    atomicAdd(&bnStats[320 + threadIdx.x], csq[threadIdx.x]);
  }
}

// ---------------- K10: finalize BN2 ----------------
__global__ void k_finalize2(float* stats, const float* g2, const float* b2) {
  int t = threadIdx.x;
  if (t < 64) {
    float mu = stats[256 + t] / (float)N_;
    float var = stats[320 + t] / (float)N_ - mu * mu;
    float sc = rsqrtf(var + 1e-5f) * g2[t];
    stats[640 + t] = sc;
    stats[704 + t] = b2[t] - mu * sc;
  }
}

// ---------------- K11: apply BN2 -> node output ----------------
__global__ void k_bn_node(const float* __restrict__ nh3, const float* __restrict__ stats,
                          float* outN) {
  size_t i = (size_t)blockIdx.x * 256 + threadIdx.x;   // grid == N*16 exactly
  size_t base = i * 4;
  int col = (int)(base & 63);
  float4 v = *(const float4*)(nh3 + base);
  v.x = v.x * stats[640 + col] + stats[704 + col];
  v.y = v.y * stats[640 + col + 1] + stats[704 + col + 1];
  v.z = v.z * stats[640 + col + 2] + stats[704 + col + 2];
  v.w = v.w * stats[640 + col + 3] + stats[704 + col + 3];
  *(float4*)(outN + base) = v;
}

// ---------------- host launcher ----------------
extern "C" void kernel_launch(void* const* d_in, const int* in_sizes, int n_in,
                              void* d_out, int out_size, void* d_ws, size_t ws_size,
                              hipStream_t stream) {
  (void)in_sizes; (void)n_in; (void)out_size; (void)ws_size;
  const float* x         = (const float*)d_in[0];
  const float* edge_attr = (const float*)d_in[1];
  const float* log_deg   = (const float*)d_in[2];
  const int*   eidx      = (const int*)d_in[3];
  const float* Wq  = (const float*)d_in[4];  const float* bq  = (const float*)d_in[5];
  const float* Wk  = (const float*)d_in[6];  const float* bk  = (const float*)d_in[7];
  const float* Wv  = (const float*)d_in[8];  const float* bv  = (const float*)d_in[9];
  const float* We  = (const float*)d_in[10]; const float* be  = (const float*)d_in[11];
  const float* Aw  = (const float*)d_in[12]; const float* VeRow = (const float*)d_in[13];
  const float* deg_coef = (const float*)d_in[14];
  const float* Now = (const float*)d_in[15]; const float* Nob = (const float*)d_in[16];
  const float* Eow = (const float*)d_in[17]; const float* Eob = (const float*)d_in[18];
  const float* W1  = (const float*)d_in[19]; const float* b1f = (const float*)d_in[20];
  const float* W2  = (const float*)d_in[21]; const float* b2f = (const float*)d_in[22];
  const float* g1n = (const float*)d_in[23]; const float* beta1n = (const float*)d_in[24];
  const float* g1e = (const float*)d_in[25]; const float* beta1e = (const float*)d_in[26];
  const float* g2  = (const float*)d_in[27]; const float* beta2  = (const float*)d_in[28];

  const int* srcIdx = eidx;
  const int* dstIdx = eidx + NE_;

  float* wsf = (float*)d_ws;
  float*    Qb    = wsf + OFF_Q;
  float*    Kb    = wsf + OFF_K;
  float*    Vb    = wsf + OFF_V;
  float*    score = wsf + OFF_SCORE;
  float*    aEx   = wsf + OFF_AEX;
  unsigned* amaxU = (unsigned*)(wsf + OFF_AMAX);
  float*    denom = wsf + OFF_DENOM;
  float*    aggV  = wsf + OFF_AGGV;
  float*    aggE  = wsf + OFF_AGGE;
  float*    nhRes = wsf + OFF_NHRES;
  float*    h1Buf = wsf + OFF_H1;
  float*    nh3   = wsf + OFF_NH3;
  float*    stats = wsf + OFF_STATS;

  float* outN = (float*)d_out;
  float* outE = outN + (size_t)N_ * 64;

  const int NODE_BLOCKS = (N_ + 127) / 128;    // 391
  const int EDGE_BLOCKS = NE_ / 128;           // 6250

  k_init<<<28130, 256, 0, stream>>>(wsf);
  k_qkv<<<NODE_BLOCKS, 256, 0, stream>>>(x, Wq, bq, Wk, bk, Wv, bv, Qb, Kb, Vb);
  k_edgeA<<<EDGE_BLOCKS, 256, 0, stream>>>(edge_attr, srcIdx, dstIdx, We, be, Aw,
                                           Qb, Kb, score, aEx, amaxU);
  k_softmax1<<<25000, 256, 0, stream>>>(dstIdx, amaxU, aEx, denom);
  k_scatter<<<25000, 256, 0, stream>>>(srcIdx, dstIdx, aEx, denom, Vb, score, aggV, aggE);
  k_node<<<NODE_BLOCKS, 256, 0, stream>>>(x, log_deg, aggV, aggE, VeRow, deg_coef,
                                          Now, Nob, nhRes, stats);
  k_edgeOut<<<EDGE_BLOCKS, 256, 0, stream>>>(edge_attr, score, Eow, Eob, outE, stats);
  k_finalize1<<<1, 128, 0, stream>>>(stats, g1n, beta1n, g1e, beta1e);
  k_bn_edge<<<50000, 256, 0, stream>>>(outE, stats);
  k_ffn1<<<NODE_BLOCKS, 256, 0, stream>>>(nhRes, stats, W1, b1f, h1Buf);
  k_ffn2<<<NODE_BLOCKS, 256, 0, stream>>>(h1Buf, nhRes, stats, W2, b2f, nh3, stats);
  k_finalize2<<<1, 64, 0, stream>>>(stats, g2, beta2);
  k_bn_node<<<3125, 256, 0, stream>>>(nh3, stats, outN);
}